// AZConv2d_41669772706587
// MI455X (gfx1250) — compile-verified
//
#include <hip/hip_runtime.h>
#include <math.h>
#include <stdint.h>

// Problem constants (from reference setup_inputs)
#define Bn   8
#define Cc   64
#define Hh   128
#define Ww   128
#define Ll   (Hh * Ww)          // 16384 pixels per image
#define NPIX (Bn * Ll)          // 131072 total pixels
#define COUT 64

typedef float v2f __attribute__((ext_vector_type(2)));
typedef float v8f __attribute__((ext_vector_type(8)));

__device__ __forceinline__ float softplusf(float x) {
    return (x > 20.f) ? x : log1pf(__expf(x));
}

// Async global->LDS copy (CDNA5, tracked by ASYNCcnt). Generic pointers into
// __shared__ carry the LDS byte offset in their low 32 bits (ISA 10.2), which
// is exactly the vdst operand the instruction wants.
__device__ __forceinline__ void async_copy_b32(void* lds_dst, const void* gsrc) {
    asm volatile("global_load_async_to_lds_b32 %0, %1, off"
                 :: "v"((unsigned)(uintptr_t)lds_dst),
                    "v"((unsigned long long)(uintptr_t)gsrc)
                 : "memory");
}
__device__ __forceinline__ void async_wait0() {
    asm volatile("s_wait_asynccnt 0x0" ::: "memory");
}

// ---------------------------------------------------------------------------
// Kernel 1: fused projection GEMM  [80 x 64] x [64 x Npix]
//   rows 0-3  : gate logits      -> softmax -> mu
//   rows 4-7  : theta            -> cos(2th), sin(2th)
//   rows 8-11 : raw_base         -> softplus + 1e-4
//   rows 12-15: raw_hyper        -> softplus
//   rows 16-79: value projection v
// Workspace layout (floats): v planes [64][NPIX], then field planes [20][NPIX]
//   fields: mu[4], cos2[4], sin2[4], base[4], hyper[4]  (channel-major)
// Each wave: one 16-pixel column tile, 5 M-tiles, K=64 in 16 steps of 4.
// ---------------------------------------------------------------------------
__global__ __launch_bounds__(256) void azc_proj(
    const float* __restrict__ x,
    const float* __restrict__ gate_w, const float* __restrict__ gate_b,
    const float* __restrict__ value_w, const float* __restrict__ geom_w,
    const float* __restrict__ geom_b, float* __restrict__ ws)
{
    __shared__ float Wsh[80][64];     // combined weight matrix (20 KB)
    __shared__ float Osh[80][130];    // raw GEMM outputs, padded (40.6 KB)

    const int t    = threadIdx.x;
    const int lane = t & 31;
    const int wid  = t >> 5;

    // stage combined 80x64 weights via async global->LDS DMA (no VGPR round-trip)
    for (int idx = t; idx < 80 * 64; idx += 256) {   // exactly 20 full iterations
        int m = idx >> 6, k = idx & 63;
        const float* src;
        if      (m < 4)  src = gate_w  + m * 64 + k;
        else if (m < 16) src = geom_w  + (m - 4) * 64 + k;
        else             src = value_w + (m - 16) * 64 + k;
        async_copy_b32(&Wsh[m][k], src);
    }
    async_wait0();
    __syncthreads();

    const int gpix0 = blockIdx.x * 128 + wid * 16;  // wave's 16-pixel tile
    const int b     = gpix0 >> 14;                  // / Ll (tiles never cross b)
    const int pixin = gpix0 & (Ll - 1);
    const float* xb = x + b * Cc * Ll + pixin;

    const int n    = lane & 15;               // N / M index within tile
    const int koff = ((lane >> 4) & 1) * 2;   // K split across lane halves

    v8f acc[5];
    #pragma unroll
    for (int mt = 0; mt < 5; ++mt)
        acc[mt] = (v8f){0.f, 0.f, 0.f, 0.f, 0.f, 0.f, 0.f, 0.f};

    #pragma unroll
    for (int ks = 0; ks < 16; ++ks) {
        const int k0 = ks * 4 + koff;
        v2f bv;                                   // B = x tile (k=c, n=pixel)
        bv.x = xb[k0 * Ll + n];
        bv.y = xb[(k0 + 1) * Ll + n];
        #pragma unroll
        for (int mt = 0; mt < 5; ++mt) {
            v2f av;                               // A = weights (m, k)
            av.x = Wsh[mt * 16 + n][k0];
            av.y = Wsh[mt * 16 + n][k0 + 1];
            acc[mt] = __builtin_amdgcn_wmma_f32_16x16x4_f32(
                false, av, false, bv, (short)0, acc[mt], false, false);
        }
    }

    // spill C tiles to LDS (D layout: vgpr i -> row i + 8*(lane>=16), col lane&15)
    #pragma unroll
    for (int mt = 0; mt < 5; ++mt) {
        #pragma unroll
        for (int i = 0; i < 8; ++i) {
            int row = mt * 16 + i + ((lane >> 4) & 1) * 8;
            Osh[row][wid * 16 + (lane & 15)] = acc[mt][i];
        }
    }
    __syncthreads();

    const int blk0 = blockIdx.x * 128;
    float* wsV = ws;
    float* wsF = ws + 64 * NPIX;

    // per-pixel nonlinear epilogue (128 pixels, one thread each)
    if (t < 128) {
        const int gpix = blk0 + t;
        float g[4];
        #pragma unroll
        for (int r = 0; r < 4; ++r) g[r] = Osh[r][t] + gate_b[r];
        float mx = fmaxf(fmaxf(g[0], g[1]), fmaxf(g[2], g[3]));
        float e[4]; float sm = 0.f;
        #pragma unroll
        for (int r = 0; r < 4; ++r) { e[r] = __expf(g[r] - mx); sm += e[r]; }
        float inv = 1.f / sm;
        #pragma unroll
        for (int r = 0; r < 4; ++r) {
            float th = Osh[4 + r][t] + geom_b[r];
            float s2v, c2v;
            __sincosf(2.f * th, &s2v, &c2v);
            float ba = softplusf(Osh[8 + r][t] + geom_b[4 + r]) + 1e-4f;
            float hy = softplusf(Osh[12 + r][t] + geom_b[8 + r]);
            wsF[(0  + r) * NPIX + gpix] = e[r] * inv;  // mu
            wsF[(4  + r) * NPIX + gpix] = c2v;         // cos(2 theta)
            wsF[(8  + r) * NPIX + gpix] = s2v;         // sin(2 theta)
            wsF[(12 + r) * NPIX + gpix] = ba;          // base
            wsF[(16 + r) * NPIX + gpix] = hy;          // hyper
        }
    }
    // value rows -> workspace, coalesced per channel
    for (int idx = t; idx < 64 * 128; idx += 256) {
        int c = idx >> 7, p2 = idx & 127;
        wsV[c * NPIX + blk0 + p2] = Osh[16 + c][p2];
    }
}

// ---------------------------------------------------------------------------
// Kernel 2: fused spatial aggregation + pointwise WMMA GEMM.
// One workgroup = 32 consecutive pixels of one image row.
//  - async-stage 3x34 halo of v (64 planes) and fields (20 planes) into LDS
//  - compute normalized tap weights w[pix][r][s]  (zero-pad => mu_n=0 kills taps)
//  - build agg tile [256 x 32] in two 128-row passes through LDS
//  - WMMA: pw_w [64 x 256] x agg [256 x 32], 8 waves (4 M-tiles x 2 N-tiles)
// ---------------------------------------------------------------------------
__global__ __launch_bounds__(256) void azc_spatial(
    const float* __restrict__ ws,
    const float* __restrict__ pw_w, const float* __restrict__ pw_b,
    float* __restrict__ out)
{
    __shared__ float vsh[64][104];   // v halo, 3x34 flattened, padded (26.0 KB)
    __shared__ float fsh[20][104];   // field halo (8.1 KB)
    __shared__ float wsh[32][36];    // tap weights [pix][r*9+s] (4.5 KB)
    __shared__ float ash[128][33];   // agg half-tile, padded (16.5 KB)

    const int t    = threadIdx.x;
    const int lane = t & 31;
    const int wid  = t >> 5;
    const int b = blockIdx.z, h = blockIdx.y, w0 = blockIdx.x * 32;

    // pull pw_w (64 KB, reused by all 4096 blocks) toward the caches early
    __builtin_prefetch(pw_w + t * 64, 0, 3);

    const float* wsV = ws;
    const float* wsF = ws + 64 * NPIX;

    // ---- async halo stage; OOB entries zero-filled with normal ds_store ----
    // (disjoint LDS addresses, so ASYNCcnt/DScnt cross-ordering is irrelevant;
    //  tail-iteration EXEC masking just suppresses the per-lane async op)
    for (int idx = t; idx < 84 * 102; idx += 256) {
        int plane = idx / 102;
        int rem = idx - plane * 102;
        int i = rem / 34;            // halo row 0..2  (h-1..h+1)
        int j = rem - i * 34;        // halo col 0..33 (w0-1..w0+32)
        int hh = h - 1 + i;
        int wwc = w0 - 1 + j;
        bool ok = ((unsigned)hh < (unsigned)Hh) && ((unsigned)wwc < (unsigned)Ww);
        int g = b * Ll + hh * Ww + wwc;
        float* dst;
        const float* src;
        if (plane < 64) { dst = &vsh[plane][rem];      src = wsV + plane * NPIX + g; }
        else            { dst = &fsh[plane - 64][rem]; src = wsF + (plane - 64) * NPIX + g; }
        if (ok) async_copy_b32(dst, src);
        else    *dst = 0.f;
    }
    async_wait0();
    __syncthreads();

    // ---- phase 1: normalized tap weights (one thread per (pix, r)) ----
    if (t < 128) {
        const int pix = t >> 2, r = t & 3;
        const int ic = 34 + 1 + pix;                 // center (row 1, col 1+pix)
        const float cc = fsh[4 + r][ic],  sc = fsh[8 + r][ic];
        const float bc = fsh[12 + r][ic], hc = fsh[16 + r][ic];
        float cmp[9]; float sum = 0.f;
        #pragma unroll
        for (int s = 0; s < 9; ++s) {
            const int dy = s / 3 - 1, dx = s - (s / 3) * 3 - 1;
            const int off = (1 + dy) * 34 + (1 + dx + pix);
            // doubled-angle symmetrization, half-angle recovery (no atan2)
            float c2 = cc + fsh[4 + r][off];
            float s2 = sc + fsh[8 + r][off];
            float nrm = fmaxf(sqrtf(c2 * c2 + s2 * s2), 1e-6f);
            float c2p = c2 / nrm, s2p = s2 / nrm;
            float ux = sqrtf(fmaxf(0.f, 0.5f * (1.f + c2p)));
            float uy = (ux > 1e-4f) ? (s2p / (2.f * ux))
                                    : ((s2p >= 0.f) ? 1.f : -1.f);
            float bp = 0.5f * (bc + fsh[12 + r][off]);
            float hp = 0.5f * (hc + fsh[16 + r][off]);
            float e  = __expf(hp);
            float su = bp * e, ss = bp / e;
            float pu =  ux * (float)dx + uy * (float)dy;
            float ps = -uy * (float)dx + ux * (float)dy;
            float kern = __expf(-(pu * pu) / (su * su) - (ps * ps) / (ss * ss));
            float cm = kern * fsh[r][off];           // * mu_n (0 at OOB taps)
            cmp[s] = cm; sum += cm;
        }
        float inv = 1.f / (sum + 1e-6f);
        #pragma unroll
        for (int s = 0; s < 9; ++s) wsh[pix][r * 9 + s] = cmp[s] * inv;
    }
    __syncthreads();

    // ---- phases 2+3: agg tile (two 128-row passes) + pointwise WMMA ----
    v8f accc = (v8f){0.f, 0.f, 0.f, 0.f, 0.f, 0.f, 0.f, 0.f};
    const int mt   = wid >> 1;                 // M tile 0..3
    const int nt   = wid & 1;                  // N tile 0..1
    const int mrow = mt * 16 + (lane & 15);    // pw_w output row for this lane
    const int koff = ((lane >> 4) & 1) * 2;
    const int ncol = nt * 16 + (lane & 15);

    for (int p = 0; p < 2; ++p) {
        // build rows [p*128, p*128+128) of agg: agg[rc][pix] = sum_s w * v_nbr
        {
            const int rr = t >> 1, half = t & 1;
            const int rc = p * 128 + rr;
            const int r = rc >> 6, c = rc & 63;
            for (int j = 0; j < 16; ++j) {
                const int pix = half * 16 + j;
                float a = 0.f;
                #pragma unroll
                for (int s = 0; s < 9; ++s) {
                    const int dy = s / 3 - 1, dx = s - (s / 3) * 3 - 1;
                    a += wsh[pix][r * 9 + s] * vsh[c][(1 + dy) * 34 + 1 + dx + pix];
                }
                ash[rr][pix] = a;
            }
        }
        __syncthreads();
        // WMMA accumulate this K half: 32 steps of K=4
        #pragma unroll
        for (int kk = 0; kk < 32; ++kk) {
            const int k0 = kk * 4 + koff;
            const int gk = p * 128 + k0;
            v2f av, bv;
            av.x = pw_w[mrow * 256 + gk];
            av.y = pw_w[mrow * 256 + gk + 1];
            bv.x = ash[k0][ncol];
            bv.y = ash[k0 + 1][ncol];
            accc = __builtin_amdgcn_wmma_f32_16x16x4_f32(
                false, av, false, bv, (short)0, accc, false, false);
        }
        __syncthreads();
    }

    // ---- bias + store ----
    #pragma unroll
    for (int i = 0; i < 8; ++i) {
        const int row = mt * 16 + i + ((lane >> 4) & 1) * 8;
        const int col = nt * 16 + (lane & 15);
        out[(b * 64 + row) * Ll + h * Ww + w0 + col] = accc[i] + pw_b[row];
    }
}

// ---------------------------------------------------------------------------
extern "C" void kernel_launch(void* const* d_in, const int* in_sizes, int n_in,
                              void* d_out, int out_size, void* d_ws, size_t ws_size,
                              hipStream_t stream) {
    const float* x       = (const float*)d_in[0];
    const float* gate_w  = (const float*)d_in[1];
    const float* gate_b  = (const float*)d_in[2];
    const float* value_w = (const float*)d_in[3];
    const float* geom_w  = (const float*)d_in[4];
    const float* geom_b  = (const float*)d_in[5];
    const float* pw_w    = (const float*)d_in[6];
    const float* pw_b    = (const float*)d_in[7];
    float* ws  = (float*)d_ws;   // needs 84 * NPIX floats (~44 MB)
    float* out = (float*)d_out;

    azc_proj<<<NPIX / 128, 256, 0, stream>>>(x, gate_w, gate_b, value_w,
                                             geom_w, geom_b, ws);
    azc_spatial<<<dim3(Ww / 32, Hh, Bn), 256, 0, stream>>>(ws, pw_w, pw_b, out);

    (void)in_sizes; (void)n_in; (void)out_size; (void)ws_size;
}